// DeepKMeans_86878598463658
// MI455X (gfx1250) — compile-verified
//
#include <hip/hip_runtime.h>
#include <hip/hip_bf16.h>
#include <math.h>

typedef __attribute__((ext_vector_type(2))) float v2f;
typedef __attribute__((ext_vector_type(4))) float v4f;
typedef __attribute__((ext_vector_type(8))) float v8f;

#define SOFTMIN_ALPHA 1000.0f

enum { EPI_BIAS = 0, EPI_BIAS_RELU = 1, EPI_DIST = 2 };

// ---------------------------------------------------------------------------
// Register-blocked FP32 WMMA GEMM for gfx1250 (wave32).
//   C[M,N] = epilogue(A[M,K] @ B[K,N])
// 256 threads = 8 wave32s arranged WROWS x WCOLS; each wave computes a
// WTM x WTN grid of 16x16 tiles via V_WMMA_F32_16X16X4_F32.
// LDS tiles are stored k-contiguous (As[m][k], Bs[n][k], stride BK+4) so each
// WMMA fragment is one aligned ds_load_b64 directly into an even VGPR pair —
// no repacking movs. Staging uses 128-bit global loads where possible.
// BT=true reads B transposed (B stored [N,K] row-major) for cluster reps.
// ---------------------------------------------------------------------------
template <int WTM, int WTN, int WROWS, int WCOLS, bool BT, int EPI>
__global__ __launch_bounds__(256) void gemm_wmma_f32(
    const float* __restrict__ A, const float* __restrict__ B,
    const float* __restrict__ bias,   // EPI_DIST: cnorm[N]
    const float* __restrict__ rnorm,  // EPI_DIST: enorm[M], else unused
    float* __restrict__ C, int M, int N, int K)
{
    constexpr int BM = WROWS * WTM * 16;
    constexpr int BN = WCOLS * WTN * 16;
    constexpr int BK = 16;
    constexpr int LDK = BK + 4;  // padded stride (dwords): 16B-aligned rows, bank-spread
    static_assert(WROWS * WCOLS == 8, "8 waves per block");
    static_assert((BM * BK) % 1024 == 0, "A tile stages with v4f");

    __shared__ float As[BM][LDK];
    __shared__ float Bs[BN][LDK];

    const int tid  = threadIdx.x;
    const int lane = tid & 31;
    const int wave = tid >> 5;
    const int wm   = wave / WCOLS;     // wave row inside block tile
    const int wn   = wave % WCOLS;     // wave col inside block tile
    const int half = lane >> 4;        // 0: lanes 0-15, 1: lanes 16-31
    const int l16  = lane & 15;

    const int row0 = blockIdx.y * BM;
    const int col0 = blockIdx.x * BN;

    v8f acc[WTM][WTN] = {};

    for (int k0 = 0; k0 < K; k0 += BK) {
        // ---- Stage A tile: As[m][k] = A[row0+m][k0+k], 128-bit loads over k.
        #pragma unroll
        for (int l = 0; l < (BM * BK) / 1024; ++l) {
            const int i = tid + l * 256;
            const int k = (i & (BK / 4 - 1)) * 4;
            const int m = i / (BK / 4);
            const v4f t = *reinterpret_cast<const v4f*>(
                &A[(size_t)(row0 + m) * K + (k0 + k)]);
            *reinterpret_cast<v4f*>(&As[m][k]) = t;
        }
        // ---- Stage B tile: Bs[n][k] = B[k0+k][col0+n] (or transposed source).
        if constexpr (BT) {
            if constexpr ((BN * BK) % 1024 == 0) {
                #pragma unroll
                for (int l = 0; l < (BN * BK) / 1024; ++l) {
                    const int i = tid + l * 256;
                    const int k = (i & (BK / 4 - 1)) * 4;
                    const int n = i / (BK / 4);
                    const v4f t = *reinterpret_cast<const v4f*>(
                        &B[(size_t)(col0 + n) * K + (k0 + k)]);
                    *reinterpret_cast<v4f*>(&Bs[n][k]) = t;
                }
            } else {
                #pragma unroll
                for (int l = 0; l < (BN * BK) / 256; ++l) {
                    const int i = tid + l * 256;
                    const int k = i & (BK - 1);
                    const int n = i / BK;
                    Bs[n][k] = B[(size_t)(col0 + n) * K + (k0 + k)];
                }
            }
        } else {
            if constexpr ((BN * BK) % 1024 == 0) {
                // Coalesced over n; 4 scattered LDS stores per thread.
                #pragma unroll
                for (int l = 0; l < (BN * BK) / 1024; ++l) {
                    const int i = tid + l * 256;
                    const int n = (i & (BN / 4 - 1)) * 4;
                    const int k = i / (BN / 4);
                    const v4f t = *reinterpret_cast<const v4f*>(
                        &B[(size_t)(k0 + k) * N + (col0 + n)]);
                    Bs[n + 0][k] = t.x;
                    Bs[n + 1][k] = t.y;
                    Bs[n + 2][k] = t.z;
                    Bs[n + 3][k] = t.w;
                }
            } else {
                #pragma unroll
                for (int l = 0; l < (BN * BK) / 256; ++l) {
                    const int i = tid + l * 256;
                    const int n = i & (BN - 1);
                    const int k = i / BN;
                    Bs[n][k] = B[(size_t)(k0 + k) * N + (col0 + n)];
                }
            }
        }
        __syncthreads();

        // Prefetch next A tile into cache (global_prefetch_b8).
        if (k0 + BK < K)
            __builtin_prefetch(&A[(size_t)(row0 + (tid & (BM - 1))) * K + (k0 + BK)], 0, 1);

        // 4 k-steps of V_WMMA_F32_16X16X4_F32 per BK=16 chunk.
        // ISA A layout (16x4): v0 = K0 (lanes 0-15) / K2 (lanes 16-31), v1 = K1/K3.
        // ISA B layout (4x16): same K split across lane halves, N = lane&15.
        // Each fragment = one aligned ds_load_b64 (k and k+1 contiguous in LDS).
        #pragma unroll
        for (int kk = 0; kk < BK; kk += 4) {
            v2f a[WTM], b[WTN];
            #pragma unroll
            for (int i = 0; i < WTM; ++i)
                a[i] = *reinterpret_cast<const v2f*>(
                    &As[(wm * WTM + i) * 16 + l16][kk + 2 * half]);
            #pragma unroll
            for (int j = 0; j < WTN; ++j)
                b[j] = *reinterpret_cast<const v2f*>(
                    &Bs[(wn * WTN + j) * 16 + l16][kk + 2 * half]);
            #pragma unroll
            for (int i = 0; i < WTM; ++i)
                #pragma unroll
                for (int j = 0; j < WTN; ++j)
                    acc[i][j] = __builtin_amdgcn_wmma_f32_16x16x4_f32(
                        /*neg_a=*/false, a[i], /*neg_b=*/false, b[j],
                        /*c_mod=*/(short)0, acc[i][j],
                        /*reuse_a=*/false, /*reuse_b=*/false);
        }
        __syncthreads();
    }

    // Epilogue + store. C/D layout: VGPR v holds M = v + 8*half, N = lane&15.
    #pragma unroll
    for (int i = 0; i < WTM; ++i) {
        const int rbase = row0 + (wm * WTM + i) * 16 + half * 8;
        #pragma unroll
        for (int j = 0; j < WTN; ++j) {
            const int col = col0 + (wn * WTN + j) * 16 + l16;
            #pragma unroll
            for (int v = 0; v < 8; ++v) {
                const int r = rbase + v;
                float val = acc[i][j][v];
                if (EPI == EPI_BIAS_RELU) {
                    val += bias[col];
                    val = val > 0.0f ? val : 0.0f;
                } else if (EPI == EPI_BIAS) {
                    val += bias[col];
                } else { // EPI_DIST: ||e||^2 - 2 e.c + ||c||^2
                    val = rnorm[r] + bias[col] - 2.0f * val;
                }
                C[(size_t)r * N + col] = val;
            }
        }
    }
}

// Squared L2 norm of each row of X[rows, cols].
__global__ __launch_bounds__(256) void rownorm_kernel(
    const float* __restrict__ X, float* __restrict__ out, int rows, int cols)
{
    int r = blockIdx.x * 256 + threadIdx.x;
    if (r >= rows) return;
    const float* p = X + (size_t)r * cols;
    float s = 0.0f;
    #pragma unroll 8
    for (int i = 0; i < cols; ++i) {
        float v = p[i];
        s += v * v;
    }
    out[r] = s;
}

// Stable softmin-weighted distances: one block per row, K = 512, 2 elems/thread.
__global__ __launch_bounds__(256) void softmin_kernel(
    const float* __restrict__ dist, float* __restrict__ wd, int K)
{
    __shared__ float red[256];
    const int b = blockIdx.x;
    const int t = threadIdx.x;
    const float* dp = dist + (size_t)b * K;

    float d0 = dp[t];
    float d1 = dp[t + 256];

    // min-reduce
    red[t] = fminf(d0, d1);
    __syncthreads();
    for (int s = 128; s > 0; s >>= 1) {
        if (t < s) red[t] = fminf(red[t], red[t + s]);
        __syncthreads();
    }
    const float mv = red[0];
    __syncthreads();

    const float e0 = __expf(-SOFTMIN_ALPHA * (d0 - mv));
    const float e1 = __expf(-SOFTMIN_ALPHA * (d1 - mv));

    // sum-reduce
    red[t] = e0 + e1;
    __syncthreads();
    for (int s = 128; s > 0; s >>= 1) {
        if (t < s) red[t] += red[t + s];
        __syncthreads();
    }
    const float inv = 1.0f / red[0];

    float* wp = wd + (size_t)b * K;
    wp[t]       = d0 * e0 * inv;
    wp[t + 256] = d1 * e1 * inv;
}

extern "C" void kernel_launch(void* const* d_in, const int* in_sizes, int n_in,
                              void* d_out, int out_size, void* d_ws, size_t ws_size,
                              hipStream_t stream) {
    (void)in_sizes; (void)n_in; (void)out_size; (void)ws_size;

    const float* x    = (const float*)d_in[0];  // [B, D]
    const float* reps = (const float*)d_in[1];  // [K, E]
    const float* W1   = (const float*)d_in[2];  // [D, H]
    const float* b1   = (const float*)d_in[3];
    const float* W2   = (const float*)d_in[4];  // [H, E]
    const float* b2   = (const float*)d_in[5];
    const float* W3   = (const float*)d_in[6];  // [E, H]
    const float* b3   = (const float*)d_in[7];
    const float* W4   = (const float*)d_in[8];  // [H, D]
    const float* b4   = (const float*)d_in[9];

    const int B = 8192, D = 784, H = 256, E = 32, K = 512;

    // Output segments (tuple return order): weighted_dists, distances,
    // reconstruction, embeddings.
    float* out   = (float*)d_out;
    float* wd    = out;
    float* dist  = wd + (size_t)B * K;
    float* recon = dist + (size_t)B * K;
    float* emb   = recon + (size_t)B * D;

    // Workspace: h1[B*H], h2[B*H], enorm[B], cnorm[K]
    float* ws    = (float*)d_ws;
    float* h1    = ws;
    float* h2    = h1 + (size_t)B * H;
    float* enorm = h2 + (size_t)B * H;
    float* cnorm = enorm + B;

    // h1 = relu(x @ W1 + b1)        [8192,784]x[784,256]; 128x64 block, 2x2 subtiles
    gemm_wmma_f32<2, 2, 4, 2, false, EPI_BIAS_RELU>
        <<<dim3(H / 64, B / 128), 256, 0, stream>>>(x, W1, b1, nullptr, h1, B, H, D);

    // emb = h1 @ W2 + b2            [8192,256]x[256,32]; 128x32 block
    gemm_wmma_f32<2, 1, 4, 2, false, EPI_BIAS>
        <<<dim3(E / 32, B / 128), 256, 0, stream>>>(h1, W2, b2, nullptr, emb, B, E, H);

    // h2 = relu(emb @ W3 + b3)      [8192,32]x[32,256]; 128x64 block
    gemm_wmma_f32<2, 2, 4, 2, false, EPI_BIAS_RELU>
        <<<dim3(H / 64, B / 128), 256, 0, stream>>>(emb, W3, b3, nullptr, h2, B, H, E);

    // recon = h2 @ W4 + b4          [8192,256]x[256,784]; N=784=49*16 -> 256x16 block
    gemm_wmma_f32<2, 1, 8, 1, false, EPI_BIAS>
        <<<dim3(D / 16, B / 256), 256, 0, stream>>>(h2, W4, b4, nullptr, recon, B, D, H);

    // Row norms for the distance expansion.
    rownorm_kernel<<<(K + 255) / 256, 256, 0, stream>>>(reps, cnorm, K, E);
    rownorm_kernel<<<(B + 255) / 256, 256, 0, stream>>>(emb, enorm, B, E);

    // dist = enorm[b] + cnorm[k] - 2 * emb @ reps^T   [8192,32]x[32,512]^T; 128x64 block
    gemm_wmma_f32<2, 2, 4, 2, true, EPI_DIST>
        <<<dim3(K / 64, B / 128), 256, 0, stream>>>(emb, reps, cnorm, enorm, dist, B, K, E);

    // Stable softmin weighting over K=512 per row.
    softmin_kernel<<<B, 256, 0, stream>>>(dist, wd, K);
}